// LearnedRFSQE8Quantizer_43224550867082
// MI455X (gfx1250) — compile-verified
//
#include <hip/hip_runtime.h>
#include <hip/hip_bf16.h>

// Residual E8 lattice quantizer, 16 levels, N = 1M vectors of 8 f32.
// Memory-bound on the codes stream (~0.5 GB NT stores). One thread = one
// vector; all state in VGPRs. x staged block-wise into LDS via CDNA5 async
// global->LDS DMA (ASYNCcnt path); outputs via non-temporal B128 stores.

#define NLEVELS 16
#define EPSQ 1e-5f
#define CLIPV 10.0f
#define SCALEMIN 1e-6f
#define TPB 256

typedef float v4f __attribute__((ext_vector_type(4)));

// round_d8(v - shift) + shift, matching the JAX reference exactly.
__device__ __forceinline__ void round_d8_shift(const float* v, float* f, float shift) {
    float d[8];
    float sum = 0.0f;
#pragma unroll
    for (int j = 0; j < 8; ++j) {
        float t  = v[j] - shift;
        float fj = rintf(t);            // round half to even == jnp.round
        d[j] = t - fj;
        f[j] = fj;
        sum += fj;                       // exact small integer in f32
    }
    int odd = ((int)sum) & 1;            // parity (works for negatives)
    // argmax |d| , first index wins ties (strict > keeps earliest)
    float best = fabsf(d[0]);
    float bd   = d[0];
    int   bi   = 0;
#pragma unroll
    for (int j = 1; j < 8; ++j) {
        float aj = fabsf(d[j]);
        bool  gt = aj > best;
        best = gt ? aj : best;
        bd   = gt ? d[j] : bd;
        bi   = gt ? j    : bi;
    }
    float flip = (bd >= 0.0f) ? 1.0f : -1.0f;
    float fix  = odd ? flip : 0.0f;
#pragma unroll
    for (int j = 0; j < 8; ++j)          // cndmask chain, no dynamic indexing
        f[j] += ((j == bi) ? fix : 0.0f) + shift;
}

__global__ __launch_bounds__(TPB) void rfsq_e8_kernel(
    const float* __restrict__ x,            // [N,8]
    const float* __restrict__ scale_factors,// [16]
    const float* __restrict__ gammas,       // [16,8]
    const float* __restrict__ betas,        // [16,8]
    float* __restrict__ out,                // [N*8] quantized then [16*N*8] codes
    int N)
{
    __shared__ __align__(16) float tile[TPB * 8];        // 8 KB x tile
    __shared__ __align__(16) float s_scale[NLEVELS];
    __shared__ __align__(16) float s_invscale[NLEVELS];
    __shared__ __align__(16) float s_gamma[NLEVELS][8];
    __shared__ __align__(16) float s_beta[NLEVELS][8];
    __shared__ __align__(16) float s_invgamma[NLEVELS][8];

    const int tid = threadIdx.x;
    const int gid = blockIdx.x * TPB + tid;

    // ---- stage x tile via CDNA5 async global->LDS (coalesced 16B/lane) ----
    {
        const size_t nvec4  = (size_t)N * 2;                  // total float4s in x
        size_t base4 = (size_t)blockIdx.x * (TPB * 2);
        size_t i0 = base4 + (size_t)tid;
        size_t i1 = base4 + (size_t)tid + TPB;
        if (i0 >= nvec4) i0 = nvec4 - 1;                      // tail clamp
        if (i1 >= nvec4) i1 = nvec4 - 1;
        const float* g0 = x + i0 * 4;
        const float* g1 = x + i1 * 4;
        // low 32 bits of a generic pointer to __shared__ == LDS byte offset
        unsigned lds0 = (unsigned)(size_t)(void*)&tile[0];
        unsigned l0 = lds0 + (unsigned)tid * 16u;
        unsigned l1 = lds0 + ((unsigned)tid + TPB) * 16u;
        asm volatile("global_load_async_to_lds_b128 %0, %1, off"
                     :: "v"(l0), "v"(g0) : "memory");
        asm volatile("global_load_async_to_lds_b128 %0, %1, off"
                     :: "v"(l1), "v"(g1) : "memory");
    }

    // ---- per-level parameters + reciprocals, once per block ----
    if (tid < NLEVELS * 8) {
        int l = tid >> 3, j = tid & 7;
        float g = gammas[l * 8 + j];
        s_gamma[l][j]    = g;
        s_invgamma[l][j] = 1.0f / g;
        s_beta[l][j]     = betas[l * 8 + j];
        if (j == 0) {
            float sc = fmaxf(scale_factors[l], SCALEMIN);    // clip(scale, min)
            s_scale[l]    = sc;
            s_invscale[l] = 1.0f / sc;
        }
    }

    asm volatile("s_wait_asynccnt 0" ::: "memory");           // this wave's DMA done
    __syncthreads();                                          // all waves' tiles visible

    if (gid >= N) return;

    float xv[8], r[8], qsum[8];
#pragma unroll
    for (int j = 0; j < 8; ++j) {
        xv[j]   = tile[tid * 8 + j];
        r[j]    = xv[j];
        qsum[j] = 0.0f;
    }

    const size_t qoff      = (size_t)gid * 8;
    float* __restrict__ codes = out + (size_t)N * 8;

    for (int l = 0; l < NLEVELS; ++l) {
        float scale     = s_scale[l];
        float inv_scale = s_invscale[l];

        // LayerNorm stats (two-pass, matches jnp.mean / jnp.var)
        float mean = 0.0f;
#pragma unroll
        for (int j = 0; j < 8; ++j) mean += r[j];
        mean *= 0.125f;
        float var = 0.0f;
#pragma unroll
        for (int j = 0; j < 8; ++j) { float d = r[j] - mean; var += d * d; }
        var *= 0.125f;
        float std     = sqrtf(var + EPSQ);
        float inv_std = 1.0f / std;

        float s[8];
#pragma unroll
        for (int j = 0; j < 8; ++j)
            s[j] = ((r[j] - mean) * inv_std * s_gamma[l][j] + s_beta[l][j]) * scale;

        // nearest E8 = argmin over {D8, D8 + 1/2}
        float a[8], b[8];
        round_d8_shift(s, a, 0.0f);
        round_d8_shift(s, b, 0.5f);
        float da = 0.0f, db = 0.0f;
#pragma unroll
        for (int j = 0; j < 8; ++j) {
            float ea = s[j] - a[j]; da += ea * ea;
            float eb = s[j] - b[j]; db += eb * eb;
        }
        bool useA = (da <= db);

        float code[8];
#pragma unroll
        for (int j = 0; j < 8; ++j) {
            float q = useA ? a[j] : b[j];
            code[j] = rintf(2.0f * q);                        // exact half-step ints
            float y = (q * inv_scale - s_beta[l][j]) * s_invgamma[l][j] * std + mean;
            qsum[j] += y;
            r[j] = fminf(fmaxf(xv[j] - qsum[j], -CLIPV), CLIPV);
        }

        // NT streaming store of codes[l, gid, 0:8] (never re-read, > L2)
        v4f c0 = { code[0], code[1], code[2], code[3] };
        v4f c1 = { code[4], code[5], code[6], code[7] };
        float* cp = codes + (size_t)l * (size_t)N * 8 + qoff;
        __builtin_nontemporal_store(c0, (v4f*)cp);
        __builtin_nontemporal_store(c1, (v4f*)(cp + 4));
    }

    v4f q0 = { qsum[0], qsum[1], qsum[2], qsum[3] };
    v4f q1 = { qsum[4], qsum[5], qsum[6], qsum[7] };
    __builtin_nontemporal_store(q0, (v4f*)(out + qoff));
    __builtin_nontemporal_store(q1, (v4f*)(out + qoff + 4));
}

extern "C" void kernel_launch(void* const* d_in, const int* in_sizes, int n_in,
                              void* d_out, int out_size, void* d_ws, size_t ws_size,
                              hipStream_t stream) {
    const float* x   = (const float*)d_in[0];   // [N,8]
    const float* sf  = (const float*)d_in[1];   // [16]
    const float* gam = (const float*)d_in[2];   // [16,8]
    const float* bet = (const float*)d_in[3];   // [16,8]
    const int N      = in_sizes[0] / 8;
    const int blocks = (N + TPB - 1) / TPB;
    rfsq_e8_kernel<<<blocks, TPB, 0, stream>>>(x, sf, gam, bet, (float*)d_out, N);
}